// Homogenizer_8675833938583
// MI455X (gfx1250) — compile-verified
//
#include <hip/hip_runtime.h>
#include <hip/hip_bf16.h>
#include <stdint.h>

// Problem constants (match reference/setup_inputs)
#define BATCH   8
#define HGT     1024
#define WID     2048
#define HW      (HGT * WID)        // 2,097,152 pixels per image (2^21)
#define NREG    2048               // N_REGIONS
#define NCLS    6                  // CLASS_NUM
#define RC      (NREG * NCLS)      // 12,288 histogram bins per image
#define IGNORE_L 255
#define PCT      0.9f

// ---------------------------------------------------------------------------
// CDNA5 async global->LDS path (ASYNCcnt). Guarded so compilation never breaks
// if the toolchain spells these differently; fallback = plain loads.
// ---------------------------------------------------------------------------
#if defined(__has_builtin)
#if __has_builtin(__builtin_amdgcn_global_load_async_to_lds_b64) && \
    __has_builtin(__builtin_amdgcn_s_wait_asynccnt)
#define USE_ASYNC_LDS 1
#endif
#endif
#ifndef USE_ASYNC_LDS
#define USE_ASYNC_LDS 0
#endif

typedef int v2i __attribute__((vector_size(2 * sizeof(int))));

#if USE_ASYNC_LDS
// clang prints AS1 as "__device__" and AS3 as "__shared__" in HIP diagnostics;
// the builtin's params are v2i pointers in those address spaces.
typedef __attribute__((address_space(1))) v2i gv2i_t;
typedef __attribute__((address_space(3))) v2i lv2i_t;
#define WAIT_ASYNC(n) __builtin_amdgcn_s_wait_asynccnt(n)
#else
#define WAIT_ASYNC(n) ((void)0)
#endif

__device__ __forceinline__ void async_cp8(const int* __restrict__ gsrc, void* ldst) {
#if USE_ASYNC_LDS
    // global ptr -> AS1 via inttoptr (AS1 addresses == generic addresses for
    // global memory); LDS generic ptr -> AS3 via explicit addrspacecast.
    __builtin_amdgcn_global_load_async_to_lds_b64(
        (gv2i_t*)(uintptr_t)gsrc,
        (lv2i_t*)ldst,
        0, 0);
#else
    *(v2i*)ldst = *(const v2i*)gsrc;
#endif
}

// ---------------------------------------------------------------------------
// Phase 1: per-block LDS histogram of (region, class) counts, flushed with
// global u32 atomics into counts[b][r][c].
// Grid: BATCH*32 blocks of 256 threads; each block owns a 65,536-pixel slice.
// ---------------------------------------------------------------------------
__global__ void __launch_bounds__(256)
hom_hist_kernel(const int* __restrict__ labels,
                const int* __restrict__ regions,
                unsigned int* __restrict__ counts) {
    constexpr int THREADS = 256;
    constexpr int PPB     = 65536;          // pixels per block
    constexpr int CHUNK   = THREADS * 2;    // 512 pixels per double-buffer step
    constexpr int NCHUNK  = PPB / CHUNK;    // 128

    __shared__ unsigned int hist[RC];       // 48 KB
    __shared__ int2 tlab[2][THREADS];       // 4 KB (double-buffered label tile)
    __shared__ int2 treg[2][THREADS];       // 4 KB (double-buffered region tile)

    const int t = threadIdx.x;
    for (int i = t; i < RC; i += THREADS) hist[i] = 0u;
    __syncthreads();

    const int img   = blockIdx.x >> 5;      // 32 slices per image
    const int slice = blockIdx.x & 31;
    const long long base = (long long)img * HW + (long long)slice * PPB;
    const int* lab = labels  + base;
    const int* reg = regions + base;

    auto issue = [&](int c, int buf) {
        const int off = c * CHUNK + t * 2;
        async_cp8(lab + off, &tlab[buf][t]);
        async_cp8(reg + off, &treg[buf][t]);
    };
    auto process = [&](int buf) {
        const int2 L = tlab[buf][t];
        const int2 R = treg[buf][t];
        if ((unsigned)L.x < (unsigned)NCLS)
            atomicAdd(&hist[(R.x & (NREG - 1)) * NCLS + L.x], 1u);
        if ((unsigned)L.y < (unsigned)NCLS)
            atomicAdd(&hist[(R.y & (NREG - 1)) * NCLS + L.y], 1u);
    };

    issue(0, 0);
    for (int c = 0; c < NCHUNK - 1; ++c) {
        issue(c + 1, (c + 1) & 1);
        WAIT_ASYNC(2);                 // oldest pair (chunk c) complete
        process(c & 1);
    }
    WAIT_ASYNC(0);
    process((NCHUNK - 1) & 1);

    __syncthreads();
    unsigned int* cdst = counts + (size_t)img * RC;
    for (int i = t; i < RC; i += THREADS) {
        unsigned int v = hist[i];
        if (v) atomicAdd(&cdst[i], v);
    }
}

// ---------------------------------------------------------------------------
// Phase 2: per (image, region) decision. 16,384 threads total.
// ---------------------------------------------------------------------------
__global__ void __launch_bounds__(256)
hom_decide_kernel(const unsigned int* __restrict__ counts,
                  int* __restrict__ decision) {
    const int idx = blockIdx.x * 256 + threadIdx.x;   // 0 .. BATCH*NREG-1
    if (idx >= BATCH * NREG) return;
    const unsigned int* c = counts + (size_t)idx * NCLS;
    unsigned int total = 0, maxc = 0;
    int amax = 0;
#pragma unroll
    for (int k = 0; k < NCLS; ++k) {
        const unsigned int v = c[k];
        total += v;
        if (v > maxc) { maxc = v; amax = k; }   // strict > keeps FIRST max (jnp.argmax)
    }
    const float pct = (float)maxc / ((float)total + 1e-5f);
    decision[idx] = (pct < PCT) ? IGNORE_L : amax;
}

// ---------------------------------------------------------------------------
// Phase 3: gather decisions back to pixels; region 0 and IGNORE fall back to
// the original pseudo label. 4 pixels per thread, b128 loads / b128 store.
// ---------------------------------------------------------------------------
__global__ void __launch_bounds__(256)
hom_scatter_kernel(const int* __restrict__ labels,
                   const int* __restrict__ regions,
                   const int* __restrict__ decision,
                   float* __restrict__ out) {
    const long long tid = (long long)blockIdx.x * 256 + threadIdx.x;
    const long long pix = tid * 4;                     // HW % 4 == 0
    const int img = (int)(pix >> 21);                  // HW == 2^21
    const int4 L = *(const int4*)(labels  + pix);
    const int4 R = *(const int4*)(regions + pix);
    const int* dec = decision + (size_t)img * NREG;

    auto resolve = [&](int l, int r) -> float {
        const int d = (r == 0) ? IGNORE_L : dec[r & (NREG - 1)];
        return (float)((d == IGNORE_L) ? l : d);
    };
    float4 o;
    o.x = resolve(L.x, R.x);
    o.y = resolve(L.y, R.y);
    o.z = resolve(L.z, R.z);
    o.w = resolve(L.w, R.w);
    *(float4*)(out + pix) = o;
}

// ---------------------------------------------------------------------------
// Host launcher. Workspace layout:
//   [0 .. 393215]        counts   (BATCH*NREG*NCLS u32)
//   [393216 .. 458751]   decision (BATCH*NREG i32)
// ---------------------------------------------------------------------------
extern "C" void kernel_launch(void* const* d_in, const int* in_sizes, int n_in,
                              void* d_out, int out_size, void* d_ws, size_t ws_size,
                              hipStream_t stream) {
    const int* labels  = (const int*)d_in[0];
    const int* regions = (const int*)d_in[1];
    float*     out     = (float*)d_out;

    unsigned int* counts   = (unsigned int*)d_ws;
    int*          decision = (int*)((char*)d_ws + (size_t)BATCH * RC * sizeof(unsigned int));

    // counts must start at zero every call (harness does not re-poison between replays)
    (void)hipMemsetAsync(counts, 0, (size_t)BATCH * RC * sizeof(unsigned int), stream);

    hom_hist_kernel<<<dim3(BATCH * 32), dim3(256), 0, stream>>>(labels, regions, counts);
    hom_decide_kernel<<<dim3((BATCH * NREG) / 256), dim3(256), 0, stream>>>(counts, decision);
    hom_scatter_kernel<<<dim3((BATCH * (long long)HW / 4) / 256), dim3(256), 0, stream>>>(
        labels, regions, decision, out);
}